// QuaternionConcat_16131897164088
// MI455X (gfx1250) — compile-verified
//
#include <hip/hip_runtime.h>

typedef float v2f __attribute__((ext_vector_type(2)));
typedef float v8f __attribute__((ext_vector_type(8)));

#define B_      8
#define HH      80
#define WW      80
#define HWH     6400        // 80*80
#define NPOS    51200       // B_*HWH
#define CIN_Q   192
#define KDIM    768         // 192*4
#define MDIM    256         // 64*4
#define LDS_PAD 16          // float2 pad (128B) between the two K-halves -> conflict-free ds_load_b64
#define SLICE_HALF (MDIM + LDS_PAD)   // 272

// W_big[m = 4*o + p][k = 4*c + q] with Hamilton sign structure
__device__ __forceinline__ float wbig_elem(int m, int k,
    const float* __restrict__ Wr, const float* __restrict__ Wi,
    const float* __restrict__ Wj, const float* __restrict__ Wk) {
  int o = m >> 2, p = m & 3;
  int c = k >> 2, q = k & 3;
  int wi = o * CIN_Q + c;
  float r = Wr[wi], i = Wi[wi], j = Wj[wi], kk = Wk[wi];
  switch (p * 4 + q) {
    case 0:  return  r;  case 1:  return -i;  case 2:  return -j;  case 3:  return -kk;
    case 4:  return  i;  case 5:  return  r;  case 6:  return -kk; case 7:  return  j;
    case 8:  return  j;  case 9:  return  kk; case 10: return  r;  case 11: return -i;
    case 12: return  kk; case 13: return -j;  case 14: return  i;  default: return  r;
  }
}

// ws layout: wpair[kk*256 + m] = { W_big[m][2kk], W_big[m][2kk+1] }  (kk in 0..383)
__global__ void __launch_bounds__(256)
build_wbig(const float* __restrict__ Wr, const float* __restrict__ Wi,
           const float* __restrict__ Wj, const float* __restrict__ Wk,
           v2f* __restrict__ wpair) {
  int idx = blockIdx.x * 256 + threadIdx.x;    // 384*256 = 98304 = 384 kk * 256 m
  int m  = idx & 255;
  int kk = idx >> 8;
  int k0 = kk * 2;
  v2f w;
  w.x = wbig_elem(m, k0,     Wr, Wi, Wj, Wk);
  w.y = wbig_elem(m, k0 + 1, Wr, Wi, Wj, Wk);
  wpair[(size_t)kk * MDIM + m] = w;
}

// GEMM: y[m,n] = sum_k W_big[m,k] * x[k,n]; y stored directly in output layout
// [b][m=4o+p][hw].  Block = 4 waves, each wave: one 16-position N-tile, all 16 M-tiles.
__global__ void __launch_bounds__(128)
qgemm_wmma(const float* __restrict__ x1, const float* __restrict__ x2,
           const v2f* __restrict__ wpair, float* __restrict__ y) {
  __shared__ v2f sw[2 * SLICE_HALF];

  int tid    = threadIdx.x;
  int wave   = tid >> 5;
  int lane   = tid & 31;
  int laneM  = lane & 15;     // A-row / C-column / B-column index
  int hiHalf = lane >> 4;     // selects K pair {k0,k0+1} vs {k0+2,k0+3}

  int n0  = (blockIdx.x * 4 + wave) * 16;   // 3200 N-tiles total
  int b   = n0 / HWH;                       // tile never crosses batch (6400 % 16 == 0)
  int hw0 = n0 % HWH;
  int h   = hw0 / WW;                       // tile never crosses a row (80 % 16 == 0)
  int wc  = (hw0 % WW) + laneM;

  // x1: [8][64][4][80][80]  element (c<64, q) at offset (4c+q)*6400
  const float* px1 = x1 + (size_t)b * (256 * HWH) + h * WW + wc;
  // x2: [8][128][4][40][40] element (c>=64, q) at offset ((c-64)*4+q)*1600, nearest ups.
  const float* px2 = x2 + (size_t)b * (512 * 1600) + (h >> 1) * 40 + (wc >> 1);

  v8f zero = {};
  v8f acc[16];
#pragma unroll
  for (int i = 0; i < 16; ++i) acc[i] = zero;

  for (int ks = 0; ks < KDIM / 4; ++ks) {
    int k0 = ks * 4;

    __syncthreads();
    // stage 256x4 W slice (512 float2, contiguous) into LDS with padded halves
    const v2f* src = wpair + (size_t)ks * 512;
#pragma unroll
    for (int t = 0; t < 4; ++t) {
      int j   = tid + t * 128;                       // 0..511
      int dst = (j & 255) + (j >> 8) * SLICE_HALF;   // [khalf][m] padded
      sw[dst] = src[j];
    }
    __syncthreads();

    // B fragment: lane holds X[k0+2*hiHalf + {0,1}, n0+laneM]
    int klane = k0 + 2 * hiHalf;
    v2f bfrag;
    if (k0 < 256) {           // whole 4-group lives in x1 region (k = 4c+q)
      bfrag.x = px1[(size_t)klane * HWH];
      bfrag.y = px1[(size_t)(klane + 1) * HWH];
    } else {                  // x2 region with nearest-neighbor upsample
      bfrag.x = px2[(size_t)(klane - 256) * 1600];
      bfrag.y = px2[(size_t)(klane - 255) * 1600];
    }

    const v2f* arow = sw + hiHalf * SLICE_HALF + laneM;
#pragma unroll
    for (int mt = 0; mt < 16; ++mt) {
      v2f afrag = arow[mt * 16];   // A[m = mt*16+laneM][K pair per half]
      acc[mt] = __builtin_amdgcn_wmma_f32_16x16x4_f32(
          false, afrag, false, bfrag, (short)0, acc[mt], false, false);
    }
  }

  // D layout: VGPR r -> M = r (lanes 0-15) / r+8 (lanes 16-31), N = laneM
  size_t baseRow = (size_t)b * (256 * HWH) + hw0 + laneM;
#pragma unroll
  for (int mt = 0; mt < 16; ++mt) {
#pragma unroll
    for (int r = 0; r < 8; ++r) {
      int m = mt * 16 + r + hiHalf * 8;
      y[baseRow + (size_t)m * HWH] = acc[mt][r];
    }
  }
}

// Per-row mean / inv-std over N = 51200, fixed-order reduction (deterministic)
__global__ void __launch_bounds__(256)
iqbn_stats(const float* __restrict__ y, float* __restrict__ stats) {
  __shared__ float ssum[256], ssq[256];
  int m = blockIdx.x, tid = threadIdx.x;
  float s = 0.f, q = 0.f;
  for (int b = 0; b < B_; ++b) {
    const float* row = y + (size_t)(b * 256 + m) * HWH;
    for (int hw = tid; hw < HWH; hw += 256) {
      float v = row[hw];
      s += v; q += v * v;
    }
  }
  ssum[tid] = s; ssq[tid] = q;
  __syncthreads();
  for (int off = 128; off > 0; off >>= 1) {
    if (tid < off) { ssum[tid] += ssum[tid + off]; ssq[tid] += ssq[tid + off]; }
    __syncthreads();
  }
  if (tid == 0) {
    float mean = ssum[0] * (1.0f / NPOS);
    float var  = ssq[0] * (1.0f / NPOS) - mean * mean;
    stats[m]        = mean;
    stats[256 + m]  = rsqrtf(var + 1e-5f);
  }
}

// In-place fused (y-mean)*invstd*gamma + beta, ReLU; float4 vectorized
__global__ void __launch_bounds__(256)
iqbn_apply(float* __restrict__ y, const float* __restrict__ stats,
           const float* __restrict__ gamma, const float* __restrict__ beta) {
  size_t i = ((size_t)blockIdx.x * 256 + threadIdx.x) * 4;   // rows are multiples of 6400
  int m = (int)((i / HWH) & 255);
  float g  = gamma[m] * stats[256 + m];
  float bt = beta[m] - stats[m] * g;
  float4 v = *(const float4*)(y + i);
  v.x = fmaxf(v.x * g + bt, 0.f);
  v.y = fmaxf(v.y * g + bt, 0.f);
  v.z = fmaxf(v.z * g + bt, 0.f);
  v.w = fmaxf(v.w * g + bt, 0.f);
  *(float4*)(y + i) = v;
}

extern "C" void kernel_launch(void* const* d_in, const int* in_sizes, int n_in,
                              void* d_out, int out_size, void* d_ws, size_t ws_size,
                              hipStream_t stream) {
  const float* x1    = (const float*)d_in[0];
  const float* x2    = (const float*)d_in[1];
  const float* Wr    = (const float*)d_in[2];
  const float* Wi    = (const float*)d_in[3];
  const float* Wj    = (const float*)d_in[4];
  const float* Wk    = (const float*)d_in[5];
  const float* gamma = (const float*)d_in[6];
  const float* beta  = (const float*)d_in[7];
  float* out = (float*)d_out;

  v2f*   wpair = (v2f*)d_ws;                               // 384*256 float2 = 768 KB
  float* stats = (float*)((char*)d_ws + 768 * 1024);       // 512 floats

  build_wbig<<<384, 256, 0, stream>>>(Wr, Wi, Wj, Wk, wpair);
  qgemm_wmma<<<800, 128, 0, stream>>>(x1, x2, wpair, out); // 3200 N-tiles / 4 waves
  iqbn_stats<<<256, 256, 0, stream>>>(out, stats);
  iqbn_apply<<<NPOS * MDIM / 4 / 256, 256, 0, stream>>>(out, stats, gamma, beta);
}